// AttentionCell_78752520340006
// MI455X (gfx1250) — compile-verified
//
#include <hip/hip_runtime.h>
#include <hip/hip_bf16.h>

// Problem sizes (fixed by the reference)
#define BB   32
#define TT   512
#define UU   1024
#define EE   1024
#define DIN  256
#define KTOT (DIN + EE + UU)   // 2304 stacked K for the fused z-GEMM
#define N4U  (4 * UU)          // 4096

typedef __attribute__((ext_vector_type(2))) float v2f;
typedef __attribute__((ext_vector_type(8))) float v8f;

// CDNA5 WMMA: D = A(16x4 f32) * B(4x16 f32) + C(16x16 f32)
__device__ __forceinline__ v8f wmma4(v2f a, v2f b, v8f c) {
  return __builtin_amdgcn_wmma_f32_16x16x4_f32(
      /*neg_a=*/false, a, /*neg_b=*/false, b,
      /*c_mod=*/(short)0, c, /*reuse_a=*/false, /*reuse_b=*/false);
}

__device__ __forceinline__ float sigmoidf_(float x) {
  return 1.0f / (1.0f + __expf(-x));
}

// ---------------------------------------------------------------------------
// K1: query[b,u] = h[b,:] @ Wa_w[:,u] + Wa_b[u]    (M=32, N=1024, K=1024)
// block = 128 threads (4 waves); wave -> one 16x16 tile; grid = (N/64, M/16)
// ---------------------------------------------------------------------------
__global__ __launch_bounds__(128) void gemm_query_kernel(
    const float* __restrict__ A,   // h [32,1024]
    const float* __restrict__ Bm,  // Wa_w [1024,1024]
    const float* __restrict__ bias,// Wa_b [1024]
    float* __restrict__ D) {       // query [32,1024]
  const int N = UU, K = UU;
  int wave = threadIdx.x >> 5, lane = threadIdx.x & 31;
  int n0 = (blockIdx.x * 4 + wave) * 16;
  int m0 = blockIdx.y * 16;
  int l15 = lane & 15;
  int kh = (lane >> 4) << 1;                 // lanes 16-31 hold K+2,K+3
  const float* Arow = A + (size_t)(m0 + l15) * K;
  const float* Bcol = Bm + n0 + l15;
  v8f acc = {};
  for (int k = 0; k < K; k += 4) {
    v2f a = *(const v2f*)(Arow + k + kh);
    v2f b;
    b.x = Bcol[(size_t)(k + kh) * N];
    b.y = Bcol[(size_t)(k + kh + 1) * N];
    acc = wmma4(a, b, acc);
  }
  int col = n0 + l15;
  int rb = m0 + ((lane >> 4) << 3);          // lanes 16-31 -> rows M+8..M+15
  float bs = bias[col];
#pragma unroll
  for (int r = 0; r < 8; ++r)
    D[(size_t)(rb + r) * N + col] = acc[r] + bs;
}

// ---------------------------------------------------------------------------
// K2: logits[b,t] = sum_u tanh(query[b,u] + enc[b,t,u]) * va_w[u] + va_b
// one wave per (b,t) row; streams encodestate (64 MB) coalesced as float4
// grid = B*T/8 blocks of 256 threads (8 waves)
// ---------------------------------------------------------------------------
__global__ __launch_bounds__(256) void attn_logits_kernel(
    const float* __restrict__ query, const float* __restrict__ enc,
    const float* __restrict__ va_w, const float* __restrict__ va_b,
    float* __restrict__ logits) {
  int wave = threadIdx.x >> 5, lane = threadIdx.x & 31;
  int row = blockIdx.x * 8 + wave;           // row = b*T + t
  int b = row >> 9;                          // T = 512
  const float4* e4 = (const float4*)(enc + (size_t)row * UU);
  const float4* q4 = (const float4*)(query + (size_t)b * UU);
  const float4* v4 = (const float4*)va_w;
  float acc = 0.0f;
#pragma unroll 2
  for (int i = lane; i < UU / 4; i += 32) {
    float4 e = e4[i], q = q4[i], v = v4[i];
    acc += tanhf(q.x + e.x) * v.x + tanhf(q.y + e.y) * v.y +
           tanhf(q.z + e.z) * v.z + tanhf(q.w + e.w) * v.w;
  }
#pragma unroll
  for (int off = 16; off > 0; off >>= 1) acc += __shfl_down(acc, off);
  if (lane == 0) logits[row] = acc + va_b[0];
}

// ---------------------------------------------------------------------------
// K3: softmax over T=512 per batch row, in place. 32 blocks of 256 threads.
// ---------------------------------------------------------------------------
__global__ __launch_bounds__(256) void softmax_kernel(float* __restrict__ attn) {
  __shared__ float red[256];
  int tid = threadIdx.x;
  float* row = attn + (size_t)blockIdx.x * TT;
  float x0 = row[tid], x1 = row[tid + 256];
  red[tid] = fmaxf(x0, x1);
  __syncthreads();
  for (int s = 128; s > 0; s >>= 1) {
    if (tid < s) red[tid] = fmaxf(red[tid], red[tid + s]);
    __syncthreads();
  }
  float m = red[0];
  __syncthreads();
  float e0 = __expf(x0 - m), e1 = __expf(x1 - m);
  red[tid] = e0 + e1;
  __syncthreads();
  for (int s = 128; s > 0; s >>= 1) {
    if (tid < s) red[tid] += red[tid + s];
    __syncthreads();
  }
  float inv = 1.0f / red[0];
  row[tid] = e0 * inv;
  row[tid + 256] = e1 * inv;
}

// ---------------------------------------------------------------------------
// K4: partial context: for t-chunk ch, partial[ch,b,e] = sum_{t in chunk}
//     attn[b,t] * speech[b,t,e].  Streams speech_encode (64 MB) coalesced.
// grid = (B, 8) blocks of 256 threads; each thread owns 4 e-values.
// ---------------------------------------------------------------------------
__global__ __launch_bounds__(256) void context_partial_kernel(
    const float* __restrict__ attn, const float* __restrict__ speech,
    float* __restrict__ partial) {
  int b = blockIdx.x, ch = blockIdx.y, tid = threadIdx.x;
  const float* arow = attn + (size_t)b * TT + ch * 64;
  const float4* sp =
      (const float4*)(speech + ((size_t)b * TT + (size_t)ch * 64) * EE);
  float4 acc = {0.f, 0.f, 0.f, 0.f};
  for (int t = 0; t < 64; ++t) {
    float a = arow[t];
    float4 s = sp[(size_t)t * (EE / 4) + tid];
    acc.x += a * s.x; acc.y += a * s.y; acc.z += a * s.z; acc.w += a * s.w;
  }
  ((float4*)partial)[((size_t)ch * BB + b) * (EE / 4) + tid] = acc;
}

// ---------------------------------------------------------------------------
// K5a: assemble x_full[b, 0:2304] = [inputs | reduce(partial context) | h]
// grid covers 32*2304 elements.
// ---------------------------------------------------------------------------
__global__ __launch_bounds__(256) void assemble_x_kernel(
    const float* __restrict__ inputs, const float* __restrict__ h,
    const float* __restrict__ partial, float* __restrict__ xfull) {
  int idx = blockIdx.x * 256 + threadIdx.x;  // 0 .. 32*2304-1
  int b = idx / KTOT, k = idx - b * KTOT;
  float v;
  if (k < DIN) {
    v = inputs[b * DIN + k];
  } else if (k < DIN + EE) {
    int e = k - DIN;
    v = 0.0f;
#pragma unroll
    for (int ch = 0; ch < 8; ++ch)
      v += partial[((size_t)ch * BB + b) * EE + e];
  } else {
    v = h[b * UU + (k - DIN - EE)];
  }
  xfull[idx] = v;
}

// ---------------------------------------------------------------------------
// K5b: z = x_full @ [kernel; rec_kernel] + bias   (M=32, N=4096, K=2304)
// WMMA f32; K loop split over the two weight matrices (38 MB streamed once).
// block = 128 threads (4 waves); grid = (N/64=64, M/16=2) -> 512 waves.
// ---------------------------------------------------------------------------
__global__ __launch_bounds__(128) void gemm_z_kernel(
    const float* __restrict__ X,   // x_full [32,2304]
    const float* __restrict__ Wk,  // kernel [1280,4096]
    const float* __restrict__ Wr,  // rec_kernel [1024,4096]
    const float* __restrict__ bias,// bias [4096]
    float* __restrict__ Z) {       // z [32,4096]
  const int N = N4U;
  int wave = threadIdx.x >> 5, lane = threadIdx.x & 31;
  int n0 = (blockIdx.x * 4 + wave) * 16;
  int m0 = blockIdx.y * 16;
  int l15 = lane & 15;
  int kh = (lane >> 4) << 1;
  const float* Arow = X + (size_t)(m0 + l15) * KTOT;
  const float* Bk = Wk + n0 + l15;
  const float* Br = Wr + n0 + l15;
  v8f acc = {};
  for (int k = 0; k < DIN + EE; k += 4) {   // x @ kernel
    v2f a = *(const v2f*)(Arow + k + kh);
    v2f b;
    b.x = Bk[(size_t)(k + kh) * N];
    b.y = Bk[(size_t)(k + kh + 1) * N];
    acc = wmma4(a, b, acc);
  }
  for (int k = 0; k < UU; k += 4) {         // h @ rec_kernel
    v2f a = *(const v2f*)(Arow + DIN + EE + k + kh);
    v2f b;
    b.x = Br[(size_t)(k + kh) * N];
    b.y = Br[(size_t)(k + kh + 1) * N];
    acc = wmma4(a, b, acc);
  }
  int col = n0 + l15;
  int rb = m0 + ((lane >> 4) << 3);
  float bs = bias[col];
#pragma unroll
  for (int r = 0; r < 8; ++r)
    Z[(size_t)(rb + r) * N + col] = acc[r] + bs;
}

// ---------------------------------------------------------------------------
// K6: LSTM gates + output. out = [h_new | h_new | c_new], each 32x1024.
// ---------------------------------------------------------------------------
__global__ __launch_bounds__(256) void lstm_gates_kernel(
    const float* __restrict__ Z, const float* __restrict__ c,
    float* __restrict__ out) {
  int i = blockIdx.x * 256 + threadIdx.x;   // 0 .. 32767
  int b = i >> 10, u = i & (UU - 1);
  const float* zr = Z + (size_t)b * N4U;
  float zi = zr[u], zf = zr[UU + u], zg = zr[2 * UU + u], zo = zr[3 * UU + u];
  float cn = sigmoidf_(zf) * c[i] + sigmoidf_(zi) * tanhf(zg);
  float hn = sigmoidf_(zo) * tanhf(cn);
  out[i] = hn;
  out[BB * UU + i] = hn;
  out[2 * BB * UU + i] = cn;
}

// ---------------------------------------------------------------------------
extern "C" void kernel_launch(void* const* d_in, const int* in_sizes, int n_in,
                              void* d_out, int out_size, void* d_ws,
                              size_t ws_size, hipStream_t stream) {
  const float* inputs     = (const float*)d_in[0];   // [32,256]
  const float* h          = (const float*)d_in[1];   // [32,1024]
  const float* c          = (const float*)d_in[2];   // [32,1024]
  const float* speech     = (const float*)d_in[3];   // [32,512,1024]
  const float* enc        = (const float*)d_in[4];   // [32,512,1024]
  const float* Wa_w       = (const float*)d_in[5];   // [1024,1024]
  const float* Wa_b       = (const float*)d_in[6];   // [1024]
  const float* va_w       = (const float*)d_in[7];   // [1024,1]
  const float* va_b       = (const float*)d_in[8];   // [1]
  const float* kernel     = (const float*)d_in[9];   // [1280,4096]
  const float* rec_kernel = (const float*)d_in[10];  // [1024,4096]
  const float* bias       = (const float*)d_in[11];  // [4096]
  float* out = (float*)d_out;

  // Workspace layout (floats): ~2.0 MB total
  float* ws      = (float*)d_ws;
  float* query   = ws;                       // 32*1024      = 32768
  float* attn    = query + BB * UU;          // 32*512       = 16384
  float* partial = attn + BB * TT;           // 8*32*1024    = 262144
  float* xfull   = partial + 8 * BB * EE;    // 32*2304      = 73728
  float* z       = xfull + BB * KTOT;        // 32*4096      = 131072

  // 1) query = h @ Wa_w + Wa_b            (WMMA f32)
  gemm_query_kernel<<<dim3(UU / 64, BB / 16), 128, 0, stream>>>(
      h, Wa_w, Wa_b, query);
  // 2) logits over encodestate (64 MB stream)
  attn_logits_kernel<<<dim3(BB * TT / 8), 256, 0, stream>>>(
      query, enc, va_w, va_b, attn);
  // 3) softmax over T (in place)
  softmax_kernel<<<dim3(BB), 256, 0, stream>>>(attn);
  // 4) context partials over speech_encode (64 MB stream)
  context_partial_kernel<<<dim3(BB, 8), 256, 0, stream>>>(
      attn, speech, partial);
  // 5a) x_full = [inputs | context | h]
  assemble_x_kernel<<<dim3(BB * KTOT / 256), 256, 0, stream>>>(
      inputs, h, partial, xfull);
  // 5b) z = x_full @ [kernel;rec_kernel] + bias   (WMMA f32, 38 MB stream)
  gemm_z_kernel<<<dim3(N4U / 64, BB / 16), 128, 0, stream>>>(
      xfull, kernel, rec_kernel, bias, z);
  // 6) gates -> (h_new, h_new, c_new)
  lstm_gates_kernel<<<dim3(BB * UU / 256), 256, 0, stream>>>(z, c, out);
}